// PITLoss_58583353917585
// MI455X (gfx1250) — compile-verified
//
#include <hip/hip_runtime.h>

// PIT loss via (x-y)^2 = x^2 + y^2 - 2xy.
// Cross term = batched Gram matrix -> V_WMMA_F32_16X16X4_F32 (fp32, K=4/instr).
// 16x16 tile packs 4 batches (M = 4 batches x 4 y-ch, N = 4 batches x 4 x-ch);
// diagonal 4x4 blocks are the per-batch Gram matrices.

typedef float v2f __attribute__((ext_vector_type(2)));
typedef float v8f __attribute__((ext_vector_type(8)));

#define T_LEN 262144
#define NBLK 256
#define WAVES_PER_GROUP (NBLK * 4)          // 4 waves/group/block
#define CHUNK (T_LEN / WAVES_PER_GROUP)     // 256 timesteps per wave (contiguous)

// ---------------------------------------------------------------------------
// Kernel 1: streaming Gram + sum-of-squares.
// Block = 256 threads = 8 waves. wave&1 selects batch-group (0: b0-3, 1: b4-7).
// Each wave: A = y rows (16 rows = 4 batches x 4 ch), B = x rows, K-loop over
// its contiguous T-chunk, 4 timesteps per v_wmma_f32_16x16x4_f32.
// A-layout (ISA 7.12.2, 32-bit A 16x4): lane L -> row m=L%16, k = v + 2*(L/16)
//   => per-lane contiguous float2 load.  B (4x16) is the mirrored layout.
// ---------------------------------------------------------------------------
__global__ __launch_bounds__(256) void pit_gram(const float* __restrict__ x,
                                                const float* __restrict__ y,
                                                float* __restrict__ wsG,
                                                float* __restrict__ wsS) {
  const int lane  = threadIdx.x & 31;
  const int wave  = threadIdx.x >> 5;
  const int g     = wave & 1;                       // batch group
  const int slice = blockIdx.x * 4 + (wave >> 1);   // T-slice within group
  const int m     = lane & 15;                      // matrix row/col index
  const int h     = lane >> 4;                      // lane half -> k base

  const size_t rowbase = (size_t)(g * 16 + m) * (size_t)T_LEN;
  const float* __restrict__ yrow = y + rowbase;
  const float* __restrict__ xrow = x + rowbase;
  const size_t t0 = (size_t)slice * CHUNK + (size_t)(2 * h);

  v8f c = {0.f, 0.f, 0.f, 0.f, 0.f, 0.f, 0.f, 0.f};
  float ssx = 0.f, ssy = 0.f;

#pragma unroll 4
  for (int t = 0; t < CHUNK; t += 4) {
    const float2 av = *reinterpret_cast<const float2*>(yrow + t0 + t);
    const float2 bv = *reinterpret_cast<const float2*>(xrow + t0 + t);
    v2f a; a.x = av.x; a.y = av.y;   // A[m][k], k = 2h + {0,1}
    v2f b; b.x = bv.x; b.y = bv.y;   // B[k][n], k = 2h + {0,1}
    // D = A(16x4) x B(4x16) + C, fp32 accumulate
    c = __builtin_amdgcn_wmma_f32_16x16x4_f32(false, a, false, b,
                                              (short)0, c, false, false);
    ssy = fmaf(av.x, av.x, fmaf(av.y, av.y, ssy));  // Sum y^2 (lane pair covers k0..3)
    ssx = fmaf(bv.x, bv.x, fmaf(bv.y, bv.y, ssx));  // Sum x^2
  }

  // ---- block-level reduction of the 4 waves of each group -----------------
  __shared__ float sG[8][256];   // per wave: lane*8 + r  (C/D: M = r + 8*(lane/16), N = lane%16)
  __shared__ float sS[8][64];    // per wave: [0..31]=ssx per lane, [32..63]=ssy per lane
#pragma unroll
  for (int r = 0; r < 8; ++r) sG[wave][lane * 8 + r] = c[r];
  sS[wave][lane]      = ssx;
  sS[wave][32 + lane] = ssy;
  __syncthreads();

  const int tid = threadIdx.x;
#pragma unroll
  for (int gg = 0; gg < 2; ++gg) {
    const float v = sG[gg][tid] + sG[gg + 2][tid] + sG[gg + 4][tid] + sG[gg + 6][tid];
    wsG[((size_t)blockIdx.x * 2 + gg) * 256 + tid] = v;
    if (tid < 64) {
      const float s = sS[gg][tid] + sS[gg + 2][tid] + sS[gg + 4][tid] + sS[gg + 6][tid];
      wsS[((size_t)blockIdx.x * 2 + gg) * 64 + tid] = s;
    }
  }
}

// ---------------------------------------------------------------------------
// Kernel 2: cross-block reduction (double accumulation), assemble dists,
// brute-force the 24 permutations, emit the scalar loss.
// ---------------------------------------------------------------------------
__device__ __constant__ int kPerm[24][4] = {
  {0,1,2,3},{0,1,3,2},{0,2,1,3},{0,2,3,1},{0,3,1,2},{0,3,2,1},
  {1,0,2,3},{1,0,3,2},{1,2,0,3},{1,2,3,0},{1,3,0,2},{1,3,2,0},
  {2,0,1,3},{2,0,3,1},{2,1,0,3},{2,1,3,0},{2,3,0,1},{2,3,1,0},
  {3,0,1,2},{3,0,2,1},{3,1,0,2},{3,1,2,0},{3,2,0,1},{3,2,1,0}};

__global__ __launch_bounds__(256) void pit_finalize(const float* __restrict__ wsG,
                                                    const float* __restrict__ wsS,
                                                    float* __restrict__ out) {
  __shared__ double G[2][256];
  __shared__ double SS[2][64];
  const int tid = threadIdx.x;

#pragma unroll
  for (int g = 0; g < 2; ++g) {
    double acc = 0.0;
    for (int bk = 0; bk < NBLK; ++bk)
      acc += (double)wsG[((size_t)bk * 2 + g) * 256 + tid];
    G[g][tid] = acc;
    if (tid < 64) {
      double s = 0.0;
      for (int bk = 0; bk < NBLK; ++bk)
        s += (double)wsS[((size_t)bk * 2 + g) * 64 + tid];
      SS[g][tid] = s;
    }
  }
  __syncthreads();

  if (tid == 0) {
    const double invT = 1.0 / (double)T_LEN;
    double total = 0.0;
    for (int b = 0; b < 8; ++b) {
      const int g  = b >> 2;
      const int bl = b & 3;
      float d[4][4];
      for (int i = 0; i < 4; ++i) {
        const int M = bl * 4 + i;
        const double syy = SS[g][32 + M] + SS[g][32 + M + 16];
        for (int j = 0; j < 4; ++j) {
          const int N = bl * 4 + j;
          // C/D layout: element (M,N) lives at lane = N + 16*(M>=8), vgpr r = M&7
          const int lane2 = N + ((M >= 8) ? 16 : 0);
          const double gij = G[g][lane2 * 8 + (M & 7)];
          const double sxx = SS[g][N] + SS[g][N + 16];
          d[i][j] = (float)((syy + sxx - 2.0 * gij) * invT);
        }
      }
      float best = 3.4e38f;
      for (int p = 0; p < 24; ++p) {
        const float cst = d[0][kPerm[p][0]] + d[1][kPerm[p][1]] +
                          d[2][kPerm[p][2]] + d[3][kPerm[p][3]];
        best = fminf(best, cst);
      }
      total += (double)best;
    }
    out[0] = (float)total;
  }
}

extern "C" void kernel_launch(void* const* d_in, const int* in_sizes, int n_in,
                              void* d_out, int out_size, void* d_ws, size_t ws_size,
                              hipStream_t stream) {
  const float* x = (const float*)d_in[0];
  const float* y = (const float*)d_in[1];
  float* out = (float*)d_out;
  float* wsG = (float*)d_ws;                          // NBLK*2*256 floats
  float* wsS = wsG + (size_t)NBLK * 2 * 256;          // NBLK*2*64 floats
  pit_gram<<<NBLK, 256, 0, stream>>>(x, y, wsG, wsS);
  pit_finalize<<<1, 256, 0, stream>>>(wsG, wsS, out);
}